// GNN3D_51616916963867
// MI455X (gfx1250) — compile-verified
//
#include <hip/hip_runtime.h>
#include <math.h>

// ---------------------------------------------------------------------------
// 2-layer GCN for MI455X (gfx1250, wave32).
//   h1  = relu( A_norm @ (x @ W1) + b1 )      (transform first: 16-ch messages)
//   out = log_softmax( (A_norm @ h1) @ W2 + b2 )  (aggregate first: 16-ch msgs)
// Dense GEMMs use V_WMMA_F32_16X16X4_F32 (one wave = one 16x16 C tile).
// Aggregation is thread-per-edge: float4 gathers from L2-resident feature
// arrays + 16 global f32 atomics into L2-resident accumulators.
// ---------------------------------------------------------------------------

typedef float v2f __attribute__((ext_vector_type(2)));
typedef float v8f __attribute__((ext_vector_type(8)));

#define IN_CH  128
#define HIDDEN 16
#define OUT_CH 40

// ---- degree / normalization ----------------------------------------------

__global__ __launch_bounds__(256) void k_deg_init(float* __restrict__ deg, int n) {
    int i = blockIdx.x * 256 + threadIdx.x;
    if (i < n) deg[i] = 1.0f;               // self-loop contributes 1 to in-degree
}

__global__ __launch_bounds__(256) void k_deg_edges(const int* __restrict__ dst,
                                                   float* __restrict__ deg, int ne) {
    int e = blockIdx.x * 256 + threadIdx.x;
    if (e < ne) atomicAdd(&deg[dst[e]], 1.0f);
}

__global__ __launch_bounds__(256) void k_dis(float* __restrict__ deg, int n) {
    int i = blockIdx.x * 256 + threadIdx.x;
    if (i < n) deg[i] = rsqrtf(deg[i]);     // deg >= 1 always (self-loops)
}

// ---- GEMM1: h0 = x @ W1   [N,128] @ [128,16] -> [N,16] --------------------
// One wave per 16-row tile. A-operand (16x4 f32): lanes 0-15 hold K=0,1,
// lanes 16-31 hold K=2,3 (VGPR0=K even, VGPR1=K odd). B-operand (4x16)
// mirrors that with N striped across lanes. C/D: VGPR r -> M = r + 8*(lane/16).

__global__ __launch_bounds__(256) void k_gemm1(const float* __restrict__ x,
                                               const float* __restrict__ W1,
                                               float* __restrict__ h0,
                                               int nTiles, int n) {
    int wave = blockIdx.x * 8 + (threadIdx.x >> 5);
    int lane = threadIdx.x & 31;
    if (wave >= nTiles) return;             // wave-uniform: EXEC stays all-ones

    int half = lane >> 4;                   // 0: K={0,1}, 1: K={2,3}
    int m    = lane & 15;
    int row  = wave * 16 + m;
    if (row >= n) row = n - 1;              // clamp loads; stores guarded below

    const float4* xrow4 = (const float4*)(x + (size_t)row * IN_CH);
    v8f acc = {};
#pragma unroll
    for (int ks = 0; ks < IN_CH / 4; ++ks) {
        float4 v = xrow4[ks];               // covers K = 4*ks .. 4*ks+3
        v2f a, b;
        a.x = half ? v.z : v.x;
        a.y = half ? v.w : v.y;
        int kb = ks * 4 + half * 2;
        b.x = W1[(kb + 0) * HIDDEN + m];
        b.y = W1[(kb + 1) * HIDDEN + m];
        acc = __builtin_amdgcn_wmma_f32_16x16x4_f32(
            false, a, false, b, (short)0, acc, false, false);
    }

    int m0 = half * 8;
    int tileRow = wave * 16;
#pragma unroll
    for (int r = 0; r < 8; ++r) {
        int orow = tileRow + m0 + r;
        if (orow < n) h0[(size_t)orow * HIDDEN + m] = acc[r];
    }
}

// ---- self-loop init: agg[i][c] = dis[i]^2 * hin[i][c] ---------------------

__global__ __launch_bounds__(256) void k_selfloop(const float* __restrict__ dis,
                                                  const float* __restrict__ hin,
                                                  float* __restrict__ agg, int total) {
    int i = blockIdx.x * 256 + threadIdx.x;
    if (i < total) {
        float w = dis[i >> 4];
        agg[i] = w * w * hin[i];
    }
}

// ---- edge aggregation: agg[dst] += dis[src]*dis[dst] * hin[src] (16 ch) ---

__global__ __launch_bounds__(256) void k_edge_agg(const int* __restrict__ src,
                                                  const int* __restrict__ dst,
                                                  const float* __restrict__ dis,
                                                  const float* __restrict__ hin,
                                                  float* __restrict__ agg, int ne) {
    int e = blockIdx.x * 256 + threadIdx.x;
    if (e >= ne) return;
    __builtin_prefetch(&src[e + 4096], 0, 1);   // stream edge list ahead
    __builtin_prefetch(&dst[e + 4096], 0, 1);
    int s = src[e], d = dst[e];
    float norm = dis[s] * dis[d];
    const float4* hp = (const float4*)(hin + (size_t)s * HIDDEN);
    float* ap = agg + (size_t)d * HIDDEN;
#pragma unroll
    for (int q = 0; q < 4; ++q) {
        float4 v = hp[q];
        atomicAdd(ap + q * 4 + 0, norm * v.x);
        atomicAdd(ap + q * 4 + 1, norm * v.y);
        atomicAdd(ap + q * 4 + 2, norm * v.z);
        atomicAdd(ap + q * 4 + 3, norm * v.w);
    }
}

// ---- bias + relu in place -------------------------------------------------

__global__ __launch_bounds__(256) void k_bias_relu(float* __restrict__ h,
                                                   const float* __restrict__ b, int total) {
    int i = blockIdx.x * 256 + threadIdx.x;
    if (i < total) h[i] = fmaxf(h[i] + b[i & 15], 0.0f);
}

// ---- GEMM2: out = agg2 @ W2 + b2   [N,16] @ [16,40] -> [N,40] -------------
// One wave per 16-row tile, 3 column tiles (40 padded to 48), K=16 -> 12 WMMAs.

__global__ __launch_bounds__(256) void k_gemm2(const float* __restrict__ agg2,
                                               const float* __restrict__ W2,
                                               const float* __restrict__ b2,
                                               float* __restrict__ out,
                                               int nTiles, int n) {
    int wave = blockIdx.x * 8 + (threadIdx.x >> 5);
    int lane = threadIdx.x & 31;
    if (wave >= nTiles) return;

    int half = lane >> 4;
    int m    = lane & 15;
    int row  = wave * 16 + m;
    if (row >= n) row = n - 1;

    const float* arow = agg2 + (size_t)row * HIDDEN;
    v8f acc0 = {}, acc1 = {}, acc2 = {};
#pragma unroll
    for (int ks = 0; ks < 4; ++ks) {
        int kb = ks * 4 + half * 2;
        v2f a;
        a.x = arow[kb + 0];
        a.y = arow[kb + 1];

        int c0 = m;                         // tile 0: cols 0..15
        int c1 = 16 + m;                    // tile 1: cols 16..31
        int c2 = 32 + m;                    // tile 2: cols 32..47 (>=40 padded)
        v2f b0, b1, b2v;
        b0.x  = W2[(kb + 0) * OUT_CH + c0];
        b0.y  = W2[(kb + 1) * OUT_CH + c0];
        b1.x  = W2[(kb + 0) * OUT_CH + c1];
        b1.y  = W2[(kb + 1) * OUT_CH + c1];
        b2v.x = (c2 < OUT_CH) ? W2[(kb + 0) * OUT_CH + c2] : 0.0f;
        b2v.y = (c2 < OUT_CH) ? W2[(kb + 1) * OUT_CH + c2] : 0.0f;

        acc0 = __builtin_amdgcn_wmma_f32_16x16x4_f32(false, a, false, b0,  (short)0, acc0, false, false);
        acc1 = __builtin_amdgcn_wmma_f32_16x16x4_f32(false, a, false, b1,  (short)0, acc1, false, false);
        acc2 = __builtin_amdgcn_wmma_f32_16x16x4_f32(false, a, false, b2v, (short)0, acc2, false, false);
    }

    int m0 = half * 8;
    int tileRow = wave * 16;
#pragma unroll
    for (int r = 0; r < 8; ++r) {
        int orow = tileRow + m0 + r;
        if (orow >= n) continue;
        float* op = out + (size_t)orow * OUT_CH;
        op[m]      = acc0[r] + b2[m];
        op[16 + m] = acc1[r] + b2[16 + m];
        if (32 + m < OUT_CH) op[32 + m] = acc2[r] + b2[32 + m];
    }
}

// ---- row-wise log_softmax in place (40 channels per node) -----------------

__global__ __launch_bounds__(256) void k_logsoftmax(float* __restrict__ out, int n) {
    int i = blockIdx.x * 256 + threadIdx.x;
    if (i >= n) return;
    float* row = out + (size_t)i * OUT_CH;
    float mx = -INFINITY;
#pragma unroll
    for (int c = 0; c < OUT_CH; ++c) mx = fmaxf(mx, row[c]);
    float s = 0.0f;
#pragma unroll
    for (int c = 0; c < OUT_CH; ++c) s += expf(row[c] - mx);
    float ls = logf(s) + mx;
#pragma unroll
    for (int c = 0; c < OUT_CH; ++c) row[c] = row[c] - ls;
}

// ---------------------------------------------------------------------------

extern "C" void kernel_launch(void* const* d_in, const int* in_sizes, int n_in,
                              void* d_out, int out_size, void* d_ws, size_t ws_size,
                              hipStream_t stream) {
    const float* x  = (const float*)d_in[0];
    const int*   ei = (const int*)  d_in[1];   // edge_index [2, E]
    const float* W1 = (const float*)d_in[2];
    const float* b1 = (const float*)d_in[3];
    const float* W2 = (const float*)d_in[4];
    const float* b2 = (const float*)d_in[5];
    float* out = (float*)d_out;

    const int N = in_sizes[0] / IN_CH;
    const int E = in_sizes[1] / 2;
    const int* src = ei;
    const int* dst = ei + E;

    // workspace carve-up (256B aligned)
    char* ws = (char*)d_ws;
    size_t off = 0;
    float* dis = (float*)(ws + off); off += (((size_t)N * 4) + 255) & ~(size_t)255;
    float* h0  = (float*)(ws + off); off += (((size_t)N * HIDDEN * 4) + 255) & ~(size_t)255;
    float* h1  = (float*)(ws + off); off += (((size_t)N * HIDDEN * 4) + 255) & ~(size_t)255;
    float* ag2 = (float*)(ws + off); off += (((size_t)N * HIDDEN * 4) + 255) & ~(size_t)255;
    (void)ws_size; (void)n_in; (void)out_size;

    const int nTiles  = (N + 15) / 16;
    const int gBlocks = (nTiles + 7) / 8;      // 8 waves / block (wave32)
    const int nbN   = (N + 255) / 256;
    const int nbE   = (E + 255) / 256;
    const int nbNF  = (N * HIDDEN + 255) / 256;

    // normalization: deg (with self-loop) -> dis = deg^{-1/2}
    k_deg_init <<<nbN, 256, 0, stream>>>(dis, N);
    k_deg_edges<<<nbE, 256, 0, stream>>>(dst, dis, E);
    k_dis      <<<nbN, 256, 0, stream>>>(dis, N);

    // layer 1: h0 = x @ W1 ; h1 = relu(A_norm @ h0 + b1)
    k_gemm1    <<<gBlocks, 256, 0, stream>>>(x, W1, h0, nTiles, N);
    k_selfloop <<<nbNF, 256, 0, stream>>>(dis, h0, h1, N * HIDDEN);
    k_edge_agg <<<nbE, 256, 0, stream>>>(src, dst, dis, h0, h1, E);
    k_bias_relu<<<nbNF, 256, 0, stream>>>(h1, b1, N * HIDDEN);

    // layer 2: agg2 = A_norm @ h1 ; out = log_softmax(agg2 @ W2 + b2)
    k_selfloop <<<nbNF, 256, 0, stream>>>(dis, h1, ag2, N * HIDDEN);
    k_edge_agg <<<nbE, 256, 0, stream>>>(src, dst, dis, h1, ag2, E);
    k_gemm2    <<<gBlocks, 256, 0, stream>>>(ag2, W2, b2, out, nTiles, N);
    k_logsoftmax<<<nbN, 256, 0, stream>>>(out, N);
}